// TemporalPerception_35467839930871
// MI455X (gfx1250) — compile-verified
//
#include <hip/hip_runtime.h>
#include <hip/hip_bf16.h>

// CDNA5 / gfx1250. wave32. FP32 WMMA (V_WMMA_F32_16X16X4_F32) for the two
// small projections; everything else is bandwidth-bound streaming.

typedef float vf2 __attribute__((ext_vector_type(2)));
typedef float vf8 __attribute__((ext_vector_type(8)));

__device__ __forceinline__ vf8 wmma_f32_16x16x4(vf2 a, vf2 b, vf8 c) {
    // 8 args: (neg_a, A, neg_b, B, c_mod, C, reuse_a, reuse_b)
    return __builtin_amdgcn_wmma_f32_16x16x4_f32(false, a, false, b, (short)0, c, false, false);
}

// ---------------------------------------------------------------------------
// Kernel 1: Q[b,n] = sum_k qst[b,k] * Wq[n,k] + bq[n]     (M=B, N=512, K=512)
// One 16x16 output tile per wave; 128 f32 WMMAs per tile (K step 4).
// ---------------------------------------------------------------------------
__global__ __launch_bounds__(256) void q_proj_kernel(
    const float* __restrict__ qst, const float* __restrict__ Wq,
    const float* __restrict__ bq, float* __restrict__ Q) {
    const int C = 512;
    const int nTilesN = C / 16;                       // 32
    int wave = (blockIdx.x * blockDim.x + threadIdx.x) >> 5;
    int lane = threadIdx.x & 31;
    int m0 = (wave / nTilesN) * 16;
    int n0 = (wave % nTilesN) * 16;
    int row = lane & 15;                              // M (A) / N (B,C) index
    int hi  = lane >> 4;                              // K-half selector

    vf8 acc = {0.f, 0.f, 0.f, 0.f, 0.f, 0.f, 0.f, 0.f};
    const float* aRow = qst + (size_t)(m0 + row) * C; // A row, contiguous K
    const float* bRow = Wq  + (size_t)(n0 + row) * C; // B col = Wq row (K contiguous)
    for (int k0 = 0; k0 < C; k0 += 4) {
        int k = k0 + 2 * hi;
        vf2 a;  a.x  = aRow[k]; a.y  = aRow[k + 1];   // A[m, k], A[m, k+1]
        vf2 bf; bf.x = bRow[k]; bf.y = bRow[k + 1];   // B[k, n], B[k+1, n]
        acc = wmma_f32_16x16x4(a, bf, acc);
    }
    float bias = bq[n0 + row];
    #pragma unroll
    for (int r = 0; r < 8; ++r) {
        int m = m0 + r + 8 * hi;                      // C/D layout
        Q[(size_t)m * C + n0 + row] = acc[r] + bias;
    }
}

// ---------------------------------------------------------------------------
// Kernel 2: QW[b,h,c] = sum_d Q[b, h*128+d] * Wk[h*128+d, c]
// 4 head-block GEMMs: M=B, K=128, N=512.  QW layout: [B][4][512].
// ---------------------------------------------------------------------------
__global__ __launch_bounds__(256) void qw_proj_kernel(
    const float* __restrict__ Q, const float* __restrict__ Wk,
    float* __restrict__ QW, int B) {
    const int C = 512, HDIM = 128;
    int wave = (blockIdx.x * blockDim.x + threadIdx.x) >> 5;
    int lane = threadIdx.x & 31;
    int tilesPerHead = (B / 16) * (C / 16);
    int h   = wave / tilesPerHead;
    int rem = wave % tilesPerHead;
    int m0 = (rem / (C / 16)) * 16;
    int n0 = (rem % (C / 16)) * 16;
    int row = lane & 15;
    int hi  = lane >> 4;

    vf8 acc = {0.f, 0.f, 0.f, 0.f, 0.f, 0.f, 0.f, 0.f};
    const float* aRow  = Q  + (size_t)(m0 + row) * C + h * HDIM;
    const float* bBase = Wk + (size_t)h * HDIM * C + n0 + row;    // B[k,n]=Wk[h*128+k, n]
    for (int k0 = 0; k0 < HDIM; k0 += 4) {
        int k = k0 + 2 * hi;
        vf2 a;  a.x  = aRow[k];               a.y  = aRow[k + 1];
        vf2 bf; bf.x = bBase[(size_t)k * C];  bf.y = bBase[(size_t)(k + 1) * C];
        acc = wmma_f32_16x16x4(a, bf, acc);
    }
    #pragma unroll
    for (int r = 0; r < 8; ++r) {
        int m = m0 + r + 8 * hi;
        QW[(size_t)m * (4 * C) + h * C + n0 + row] = acc[r];
    }
}

// ---------------------------------------------------------------------------
// Kernel 3 (one block per batch b): scores -> softmax -> head-mean -> top-k
// (stable-argsort tie-break) -> gather audio/visual rows.
// ---------------------------------------------------------------------------
__global__ __launch_bounds__(256) void attn_topk_gather_kernel(
    const float* __restrict__ audio, const float* __restrict__ visual,
    const float* __restrict__ Q, const float* __restrict__ QW,
    const float* __restrict__ bk, const int* __restrict__ topk_p,
    float* __restrict__ out, int B, int T) {
    const int C = 512;
    const int QWPAD = 520;                 // 520 % 64 = 8 -> conflict-free per head
    __shared__ float qw_s[4 * QWPAD];
    __shared__ float qb_s[4];
    __shared__ float sc_s[4 * 64];
    __shared__ float w_s[64];
    __shared__ int   idx_s[16];

    int b = blockIdx.x;
    int tid = threadIdx.x;
    int K = *topk_p; if (K > 16) K = 16; if (K < 1) K = 1;

    // Stage QW[b] into LDS (padded), zero qb accumulators.
    for (int i = tid; i < 4 * C; i += 256)
        qw_s[(i >> 9) * QWPAD + (i & 511)] = QW[(size_t)b * 4 * C + i];
    if (tid < 4) qb_s[tid] = 0.0f;
    __syncthreads();

    // qb[h] = q[b,h,:] . bk[h*128 : h*128+128]  (wave shuffle reduce + LDS atomic)
    {
        const float* qrow = Q + (size_t)b * C;
        float p0 = qrow[tid]       * bk[tid];         // c = tid        -> head (tid/128)
        float p1 = qrow[tid + 256] * bk[tid + 256];   // c = tid+256    -> head +2
        #pragma unroll
        for (int off = 16; off > 0; off >>= 1) {
            p0 += __shfl_xor(p0, off, 32);
            p1 += __shfl_xor(p1, off, 32);
        }
        if ((tid & 31) == 0) {
            int w = tid >> 5;
            atomicAdd(&qb_s[(w >> 2)],     p0);
            atomicAdd(&qb_s[(w >> 2) + 2], p1);
        }
    }

    // scores[h,t] = visual[b,t,:] . QW[b,h,:]   (240 pairs, float4 streaming)
    float s = 0.0f;
    int t = tid >> 2, h = tid & 3;
    if (t < T) {
        const float4* vr = (const float4*)(visual + ((size_t)b * T + t) * C);
        const float*  qh = qw_s + h * QWPAD;
        for (int c4 = 0; c4 < C / 4; ++c4) {
            float4 v = vr[c4];
            const float* q4 = qh + (c4 << 2);
            s += v.x * q4[0] + v.y * q4[1] + v.z * q4[2] + v.w * q4[3];
        }
    }
    __syncthreads();                        // qb atomics complete
    if (t < T) sc_s[h * 64 + t] = (s + qb_s[h]) * 0.08838834764831845f; // 1/sqrt(128)
    __syncthreads();

    // Per-head softmax over T (threads 0..3).
    if (tid < 4) {
        float mx = -3.4e38f;
        for (int j = 0; j < T; ++j) mx = fmaxf(mx, sc_s[tid * 64 + j]);
        float sum = 0.0f;
        for (int j = 0; j < T; ++j) {
            float e = expf(sc_s[tid * 64 + j] - mx);
            sc_s[tid * 64 + j] = e;
            sum += e;
        }
        float inv = 1.0f / sum;
        for (int j = 0; j < T; ++j) sc_s[tid * 64 + j] *= inv;
    }
    __syncthreads();
    if (tid < T)
        w_s[tid] = 0.25f * (sc_s[tid] + sc_s[64 + tid] + sc_s[128 + tid] + sc_s[192 + tid]);
    __syncthreads();

    // Top-K: stable ascending argsort's last K == max by (value, index); '>='
    // scan keeps the larger index on ties. Then sort chosen indices ascending.
    if (tid == 0) {
        int picks[16];
        for (int j = 0; j < K; ++j) {
            float best = -1.0f; int bi = 0;
            for (int u = 0; u < T; ++u) {
                float v = w_s[u];
                if (v >= best) { best = v; bi = u; }
            }
            w_s[bi] = -2.0f;
            picks[j] = bi;
        }
        for (int j = 1; j < K; ++j) {       // insertion sort ascending
            int key = picks[j], i2 = j - 1;
            while (i2 >= 0 && picks[i2] > key) { picks[i2 + 1] = picks[i2]; --i2; }
            picks[i2 + 1] = key;
        }
        for (int j = 0; j < K; ++j) idx_s[j] = picks[j];
    }
    __syncthreads();

    // Outputs: [audio gather | visual gather | indices-as-float], flat.
    size_t outV = (size_t)B * K * C;
    size_t outI = 2 * outV;
    if (tid < K) out[outI + (size_t)b * K + tid] = (float)idx_s[tid];

    const float4* a4 = (const float4*)audio;
    const float4* v4 = (const float4*)visual;
    float4*       o4 = (float4*)out;
    int n4 = K * (C / 4);
    for (int i = tid; i < n4; i += 256) {
        int j  = i / (C / 4);
        int c4 = i % (C / 4);
        int tt = idx_s[j];
        size_t src  = ((size_t)b * T + tt) * (C / 4) + c4;
        size_t dstA = ((size_t)b * K + j) * (C / 4) + c4;
        o4[dstA]               = a4[src];
        o4[(outV >> 2) + dstA] = v4[src];
    }
}

// ---------------------------------------------------------------------------
extern "C" void kernel_launch(void* const* d_in, const int* in_sizes, int n_in,
                              void* d_out, int out_size, void* d_ws, size_t ws_size,
                              hipStream_t stream) {
    const float* audio  = (const float*)d_in[0];
    const float* visual = (const float*)d_in[1];
    const float* qst    = (const float*)d_in[2];
    const float* Wq     = (const float*)d_in[3];
    const float* bq     = (const float*)d_in[4];
    const float* Wk     = (const float*)d_in[5];
    const float* bk     = (const float*)d_in[6];
    const int*   topk   = (const int*)d_in[7];
    float* out = (float*)d_out;

    const int C = 512;
    int B = in_sizes[2] / C;              // qst is [B, C]
    int T = in_sizes[1] / (B * C);        // visual is [B, T, C]

    float* Q  = (float*)d_ws;             // B*512 floats  (4 MB)
    float* QW = Q + (size_t)B * C;        // B*4*512 floats (16 MB)

    int tiles1 = (B / 16) * (C / 16);     // 4096 waves -> 512 blocks of 8 waves
    q_proj_kernel<<<tiles1 / 8, 256, 0, stream>>>(qst, Wq, bq, Q);

    int tiles2 = 4 * (B / 16) * (C / 16); // 16384 waves -> 2048 blocks
    qw_proj_kernel<<<tiles2 / 8, 256, 0, stream>>>(Q, Wk, QW, B);

    attn_topk_gather_kernel<<<B, 256, 0, stream>>>(audio, visual, Q, QW, bk,
                                                   topk, out, B, T);
}